// ARQGPS_56040733278629
// MI455X (gfx1250) — compile-verified
//
#include <hip/hip_runtime.h>
#include <stdint.h>

// Problem constants (from reference): B=4096, L=512, N=1024, indices in {0,1}.
#define LL    512
#define NN    1024
#define SLOT_BYTES   8192u        // one step: 2*N floats (epsT[i][a][n], a-major)
#define RING_SLOTS   4u
#define ZERO_OFF     32768u       // 64B zero page (for WMMA B cols >= 2)
#define PART_OFF     32832u       // 8 waves * 16 batches * 2 floats = 1KB partials
#define SMEM_BYTES   (PART_OFF + 1024u)

typedef __attribute__((ext_vector_type(2))) float v2f;
typedef __attribute__((ext_vector_type(8))) float v8f;
typedef __attribute__((ext_vector_type(4))) unsigned int v4u;
typedef __attribute__((ext_vector_type(4))) int v4i;
typedef __attribute__((ext_vector_type(8))) int v8i;

// ---------- helpers ----------

__device__ static inline unsigned lds_byte_off(const void* p) {
    // generic -> LDS(AS3) addrspacecast; AS3 pointers are the raw LDS byte offset
    return (unsigned)(unsigned long long)(__attribute__((address_space(3))) const char*)p;
}

__device__ static inline v8f wmma_f32_16x16x4(v2f a, v2f b, v8f c) {
    // D(16x16) = A(16x4) x B(4x16) + C   (f32)
    return __builtin_amdgcn_wmma_f32_16x16x4_f32(false, a, false, b, (short)0, c,
                                                 false, false);
}

// Issue one TDM load: 8KB contiguous tile epsT[step] -> LDS ring slot (step&3).
// D# per CDNA5 ISA (1 row of 2048 x 4B elements).
__device__ static inline void tdm_load_step(const float* epsT, int step, unsigned lds_ring_base) {
    unsigned long long ga = (unsigned long long)(uintptr_t)(epsT + (size_t)step * 2048);
    v4u g0;
    g0.x = 1u;                                             // count=1, no gather
    g0.y = lds_ring_base + (unsigned)(step & 3) * SLOT_BYTES; // lds_addr
    g0.z = (unsigned)ga;                                   // global_addr[31:0]
    g0.w = (unsigned)((ga >> 32) & 0x01FFFFFFu) | 0x80000000u; // addr[56:32] | type=2
    v8i g1;
    g1[0] = 0x00020000;          // workgroup_mask=0, data_size=2 (4B)
    g1[1] = (int)(2048u << 16);  // tensor_dim0[15:0] = 2048 (bits 63:48)
    g1[2] = (int)(1u << 16);     // tensor_dim0 hi=0 | tensor_dim1 = 1
    g1[3] = (int)(2048u << 16);  // tensor_dim1 hi=0 | tile_dim0 = 2048
    g1[4] = 1;                   // tile_dim1 = 1, tile_dim2 = 0
    g1[5] = 2048;                // tensor_dim0_stride lo
    g1[6] = (int)(2048u << 16);  // stride0 hi=0 | tensor_dim1_stride lo = 2048
    g1[7] = 0;
    v4i g2 = {};                 // dims 2..4 unused (tile dims zero)
    v4i g3 = {};
#if __has_include(<hip/amd_detail/amd_gfx1250_TDM.h>)
    v8i g4 = {};
    __builtin_amdgcn_tensor_load_to_lds(g0, g1, g2, g3, g4, 0);
#else
    __builtin_amdgcn_tensor_load_to_lds(g0, g1, g2, g3, 0);
#endif
}

__device__ static inline unsigned pack4(int4 v) {
    // indices are 0/1 -> pack 4 steps into 4 bits
    return (unsigned)((v.x & 1) | ((v.y & 1) << 1) | ((v.z & 1) << 2) | ((v.w & 1) << 3));
}

// ---------- kernel 1: eps (2,N,L) -> epsT (L,2,N), coalesced LDS transpose ----------

__global__ __launch_bounds__(256) void arqgps_transpose(const float* __restrict__ in,
                                                        float* __restrict__ out) {
    __shared__ float tile[32][33];
    const int c0 = blockIdx.x * 32;   // along L (512)
    const int r0 = blockIdx.y * 32;   // along 2*N (2048), r = a*N + n
    const int tx = threadIdx.x, ty = threadIdx.y;
    #pragma unroll
    for (int rr = ty; rr < 32; rr += 8)
        tile[rr][tx] = in[(size_t)(r0 + rr) * LL + (c0 + tx)];
    __syncthreads();
    #pragma unroll
    for (int rr = ty; rr < 32; rr += 8)
        out[(size_t)(c0 + rr) * 2048 + (r0 + tx)] = tile[tx][rr];
}

// ---------- kernel 2: main autoregressive loop ----------
// Workgroup: 8 waves, 16 batches. Wave w owns n-chunk [w*128, w*128+128).
// c-state kept in VGPRs as WMMA A fragments; eps staged by TDM ring in LDS.

__global__ __launch_bounds__(256) void arqgps_main(const int* __restrict__ idx,
                                                   const float* __restrict__ epsT,
                                                   float* __restrict__ out) {
    extern __shared__ char smem[];
    const int tid  = threadIdx.x;
    const int lane = tid & 31;
    const int wave = tid >> 5;          // 0..7
    const int b16  = lane & 15;         // batch-in-tile (M) / B-matrix column
    const int hi   = lane >> 4;         // half-wave: K offset 2*hi
    const int batch = blockIdx.x * 16 + b16;
    const int chunkBase = wave * 128;
    const unsigned cb2 = (unsigned)((chunkBase + 2 * hi) * 4); // byte off within a 4KB row

    const unsigned lds0 = lds_byte_off(smem);

    // zero the zero-page (16 floats at ZERO_OFF)
    if (tid < 16) ((float*)(smem + ZERO_OFF))[tid] = 0.0f;

    // TDM prologue: preload steps 0..2 (3-deep pipeline)
    if (tid < 32) {
        tdm_load_step(epsT, 0, lds0);
        tdm_load_step(epsT, 1, lds0);
        tdm_load_step(epsT, 2, lds0);
        __builtin_amdgcn_s_wait_tensorcnt(2);   // slot 0 resident
    }
    __syncthreads();

    // per-lane spin stream for this lane's batch (halves duplicate, that's fine)
    const int* ib = idx + (size_t)batch * LL;
    int4 nxt = *(const int4*)(ib + 4);
    unsigned curPack = pack4(*(const int4*)(ib));
    int sPrev = 0;

    // c-state: 32 A-fragments, lane holds c[b16][n0 + 2*hi + {0,1}]
    float2 ca[32];
    #pragma unroll
    for (int f = 0; f < 32; ++f) { ca[f].x = 1.0f; ca[f].y = 1.0f; }

    float outAcc = 0.0f;

    for (int i = 0; i < LL; ++i) {
        if ((i & 3) == 0 && i > 0) {
            curPack = pack4(nxt);
            if (i + 4 < LL) nxt = *(const int4*)(ib + i + 4);
        }
        const int si = (int)((curPack >> (i & 3)) & 1u);

        // ---- B-fragment base: only cols 0,1 carry data, others clamp to zero page ----
        const bool act = (b16 < 2);
        const unsigned bOff  = act ? ((unsigned)(i & 3) * SLOT_BYTES +
                                      (unsigned)b16 * 4096u + cb2)
                                   : ZERO_OFF;
        const unsigned bStep = act ? 16u : 0u;

        // ---- hoist ALL B-fragment loads so they issue as one deep clause ----
        float2 bv[32];
        #pragma unroll
        for (int f = 0; f < 32; ++f)
            bv[f] = *(const float2*)(smem + bOff + bStep * (unsigned)f);

        // ---- update: c *= epsT[i-1][sPrev][n]  (slot (i-1)&3) ----
        if (i > 0) {
            const char* up = smem + ((unsigned)((i - 1) & 3) * SLOT_BYTES +
                                     (unsigned)sPrev * 4096u + cb2);
            #pragma unroll
            for (int f = 0; f < 32; ++f) {
                float2 e = *(const float2*)(up + 16 * f);
                ca[f].x *= e.x;  ca[f].y *= e.y;
            }
        }

        // ---- dots: P[b][a] += c[b][n] * epsT[i][a][n]; 4 accumulators to
        //      keep the dependent v_wmma chains short (8 deep each) ----
        v8f acc0 = {}; v8f acc1 = {}; v8f acc2 = {}; v8f acc3 = {};
        #pragma unroll
        for (int f = 0; f < 32; f += 4) {
            v2f a0; a0[0] = ca[f+0].x; a0[1] = ca[f+0].y;
            v2f b0; b0[0] = bv[f+0].x; b0[1] = bv[f+0].y;
            acc0 = wmma_f32_16x16x4(a0, b0, acc0);
            v2f a1; a1[0] = ca[f+1].x; a1[1] = ca[f+1].y;
            v2f b1; b1[0] = bv[f+1].x; b1[1] = bv[f+1].y;
            acc1 = wmma_f32_16x16x4(a1, b1, acc1);
            v2f a2; a2[0] = ca[f+2].x; a2[1] = ca[f+2].y;
            v2f b2; b2[0] = bv[f+2].x; b2[1] = bv[f+2].y;
            acc2 = wmma_f32_16x16x4(a2, b2, acc2);
            v2f a3; a3[0] = ca[f+3].x; a3[1] = ca[f+3].y;
            v2f b3; b3[0] = bv[f+3].x; b3[1] = bv[f+3].y;
            acc3 = wmma_f32_16x16x4(a3, b3, acc3);
        }

        // ---- deterministic partial export: part[wave][b][a] ----
        if (act) {
            float* part = (float*)(smem + PART_OFF + (unsigned)wave * 128u);
            #pragma unroll
            for (int r = 0; r < 8; ++r) {
                const int b = r + 8 * hi;   // C/D layout: reg r -> M=r (lo) / r+8 (hi)
                part[b * 2 + b16] = (acc0[r] + acc1[r]) + (acc2[r] + acc3[r]);
            }
        }
        __syncthreads();   // partials visible; all waves done reading ring slots

        if (tid < 32) {
            if (tid < 16) {
                // reduce 8 waves in fixed order, then logsumexp epilogue
                float P0 = 0.0f, P1 = 0.0f;
                #pragma unroll
                for (int w = 0; w < 8; ++w) {
                    float2 pv = *(const float2*)(smem + PART_OFF + (unsigned)w * 128u +
                                                 (unsigned)tid * 8u);
                    P0 += pv.x;  P1 += pv.y;
                }
                const float m = fmaxf(P0, P1);
                const float l = m + 0.5f * __logf(__expf(2.0f * (P0 - m)) +
                                                  __expf(2.0f * (P1 - m)));
                outAcc += (si ? P1 : P0) - l;
            }
            // refill ring: slot (i-1)&3 is now free -> load step i+3 into it
            if (i + 3 < LL) tdm_load_step(epsT, i + 3, lds0);
            if (i + 6 <= LL) __builtin_amdgcn_s_wait_tensorcnt(2); // step i+1 resident
            else             __builtin_amdgcn_s_wait_tensorcnt(0); // pipeline drain (tail)
        }
        sPrev = si;
        __syncthreads();   // next slot ready for everyone
    }

    if (tid < 16) out[blockIdx.x * 16 + tid] = outAcc;
}

// ---------- host-side launch ----------

extern "C" void kernel_launch(void* const* d_in, const int* in_sizes, int n_in,
                              void* d_out, int out_size, void* d_ws, size_t ws_size,
                              hipStream_t stream) {
    const int*   idx = (const int*)d_in[0];    // (B, L) int32
    const float* eps = (const float*)d_in[1];  // (2, N, L) f32
    float* epsT = (float*)d_ws;                // (L, 2, N) f32, 4 MB scratch
    float* out  = (float*)d_out;               // (B,) f32

    dim3 tg(LL / 32, (2 * NN) / 32), tb(32, 8);
    arqgps_transpose<<<tg, tb, 0, stream>>>(eps, epsT);

    arqgps_main<<<4096 / 16, 256, SMEM_BYTES, stream>>>(idx, epsT, out);
}